// FPooling_48404281426011
// MI455X (gfx1250) — compile-verified
//
#include <hip/hip_runtime.h>

// Forget-gate pooling scan: h_t = (1-f_t)*z_t + f_t*h_{t-1} along contiguous S.
// Wave32 per channel, chunked affine scan, async global->LDS double buffering.

#define WAVES_PER_WG 8
#define CHUNK 128  // 32 lanes * 4 floats per chunk per channel

typedef int v4i __attribute__((ext_vector_type(4)));
typedef __attribute__((address_space(1))) v4i as1_v4i;  // global
typedef __attribute__((address_space(3))) v4i as3_v4i;  // LDS

// ---- gfx1250 async global->LDS copy (16B per lane), ASYNCcnt tracked ----
__device__ __forceinline__ void async_copy_b128(const float* g, float* l) {
#if __has_builtin(__builtin_amdgcn_global_load_async_to_lds_b128)
    __builtin_amdgcn_global_load_async_to_lds_b128(
        (as1_v4i*)g, (as3_v4i*)l, /*offset=*/0, /*cpol=*/0);
#else
    unsigned lds_addr = (unsigned)(unsigned long long)(as3_v4i*)l;
    asm volatile("global_load_async_to_lds_b128 %0, %1, off"
                 :: "v"(lds_addr), "v"(g)
                 : "memory");
#endif
}

template <int N>
__device__ __forceinline__ void wait_asynccnt() {
    asm volatile("s_wait_asynccnt %0" :: "n"(N) : "memory");
}

// ---- wave32 lane shuffles (no LDS memory touched) ----
__device__ __forceinline__ float lane_shift_up(float x, int d) {
    int idx = (((int)(threadIdx.x & 31u)) - d) & 31;
    return __int_as_float(__builtin_amdgcn_ds_bpermute(idx << 2, __float_as_int(x)));
}

__device__ __forceinline__ float bcast_lane31(float x) {
    return __int_as_float(__builtin_amdgcn_readlane(__float_as_int(x), 31));
}

__global__ __launch_bounds__(256) void fpool_scan_kernel(
    const float* __restrict__ f, const float* __restrict__ z,
    float* __restrict__ out, int S, int channels) {
    __shared__ __align__(16) float lf[2][WAVES_PER_WG * CHUNK];
    __shared__ __align__(16) float lz[2][WAVES_PER_WG * CHUNK];

    const int lane = (int)(threadIdx.x & 31u);
    const int wave = (int)(threadIdx.x >> 5);
    const long long ch = (long long)blockIdx.x * WAVES_PER_WG + wave;
    if (ch >= channels) return;  // uniform per wave; no cross-wave barriers used

    const long long base = ch * (long long)S;
    const float* gf = f + base + 4 * lane;
    const float* gz = z + base + 4 * lane;
    float* go = out + base + 4 * lane;
    const int loff = wave * CHUNK + 4 * lane;
    const int nchunks = S / CHUNK;

    // prime the pipeline: chunk 0 -> buffer 0
    async_copy_b128(gf, &lf[0][loff]);
    async_copy_b128(gz, &lz[0][loff]);

    float h = 0.0f;  // carry h_{-1} = 0
    for (int k = 0; k < nchunks; ++k) {
        const int b = k & 1;
        if (k + 1 < nchunks) {
            // prefetch next chunk into the other buffer, then wait until only
            // those 2 ops remain outstanding (async loads complete in order)
            async_copy_b128(gf + (k + 1) * CHUNK, &lf[b ^ 1][loff]);
            async_copy_b128(gz + (k + 1) * CHUNK, &lz[b ^ 1][loff]);
            wait_asynccnt<2>();
        } else {
            wait_asynccnt<0>();
        }

        const float4 ff = *(const float4*)&lf[b][loff];  // ds_load_b128
        const float4 zz = *(const float4*)&lz[b][loff];

        // per-element additive terms g_t = (1-f_t)*z_t
        const float g0 = (1.0f - ff.x) * zz.x;
        const float g1 = (1.0f - ff.y) * zz.y;
        const float g2 = (1.0f - ff.z) * zz.z;
        const float g3 = (1.0f - ff.w) * zz.w;

        // compose this lane's 4 elements into one affine map h -> a*h + bb
        float a = ff.x, bb = g0;
        a *= ff.y; bb = fmaf(ff.y, bb, g1);
        a *= ff.z; bb = fmaf(ff.z, bb, g2);
        a *= ff.w; bb = fmaf(ff.w, bb, g3);

        // inclusive wave scan of affine maps (apply-earlier-first composition)
#pragma unroll
        for (int d = 1; d < 32; d <<= 1) {
            const float ap = lane_shift_up(a, d);
            const float bp = lane_shift_up(bb, d);
            if (lane >= d) { bb = fmaf(a, bp, bb); a *= ap; }
        }

        // exclusive prefix for this lane -> h entering this lane's segment
        float ae = lane_shift_up(a, 1);
        float be = lane_shift_up(bb, 1);
        if (lane == 0) { ae = 1.0f; be = 0.0f; }
        const float hp = fmaf(ae, h, be);

        // replay the 4 elements to get all outputs
        const float o0 = fmaf(ff.x, hp, g0);
        const float o1 = fmaf(ff.y, o0, g1);
        const float o2 = fmaf(ff.z, o1, g2);
        const float o3 = fmaf(ff.w, o2, g3);

        *(float4*)(go + k * CHUNK) = make_float4(o0, o1, o2, o3);

        // carry = full-chunk map (lane 31 inclusive prefix) applied to h
        h = bcast_lane31(fmaf(a, h, bb));
    }
}

extern "C" void kernel_launch(void* const* d_in, const int* in_sizes, int n_in,
                              void* d_out, int out_size, void* d_ws, size_t ws_size,
                              hipStream_t stream) {
    const float* f = (const float*)d_in[0];
    const float* z = (const float*)d_in[1];
    float* out = (float*)d_out;

    const int S = 2048;                       // per reference: (B,H,S)=(32,1024,2048)
    const int channels = in_sizes[0] / S;     // B*H = 32768
    const int wgs = (channels + WAVES_PER_WG - 1) / WAVES_PER_WG;

    fpool_scan_kernel<<<wgs, 256, 0, stream>>>(f, z, out, S, channels);
}